// RightLeftMaxPooling_49452253446767
// MI455X (gfx1250) — compile-verified
//
#include <hip/hip_runtime.h>

#define WAVES_PER_BLOCK 8
#define ROW_W 1024            // row width (W)
#define SEG 32                // elements per lane
#define SEG_STRIDE 36         // padded LDS stride per 32-elem segment (16B-aligned, 2-way conflicts max)
#define LDS_ROW (SEG * SEG_STRIDE)   // 1152 floats per row

__global__ __launch_bounds__(256)
void RightLeftMaxPooling_kernel(const float* __restrict__ in,
                                float* __restrict__ out,
                                int rows) {
    __shared__ float smem[WAVES_PER_BLOCK * LDS_ROW];  // 36,864 bytes

    const int lane = threadIdx.x & 31;
    const int wave = threadIdx.x >> 5;
    const int row  = blockIdx.x * WAVES_PER_BLOCK + wave;
    if (row >= rows) return;  // wave-uniform

    const unsigned waveBase = (unsigned)(wave * LDS_ROW);
    const size_t   rowOff   = (size_t)row * ROW_W;

    // ---- Async DMA: global row -> LDS (coalesced 16B/lane), padded layout ----
    const float* gin = in + rowOff;
    #pragma unroll
    for (int k = 0; k < 8; ++k) {
        const int g   = 4 * (lane + 32 * k);    // float index within row
        const int seg = g >> 5;                 // 32-elem segment id
        const int w   = g & 31;                 // offset within segment (multiple of 4)
        unsigned lds_addr =
            (unsigned)(unsigned long long)(&smem[waveBase + (unsigned)(SEG_STRIDE * seg + w)]);
        unsigned long long gaddr = (unsigned long long)(gin + g);
        asm volatile("global_load_async_to_lds_b128 %0, %1, off"
                     :: "v"(lds_addr), "v"(gaddr) : "memory");
    }
    asm volatile("s_wait_asynccnt 0" ::: "memory");

    // ---- Per-lane reverse cummax over its 32 contiguous elements ----
    const unsigned segBase = waveBase + (unsigned)(SEG_STRIDE * lane);
    float e[SEG];
    #pragma unroll
    for (int i = 0; i < SEG; ++i) e[i] = smem[segBase + i];

    float run = -__builtin_inff();
    #pragma unroll
    for (int i = SEG - 1; i >= 0; --i) { run = fmaxf(run, e[i]); e[i] = run; }

    // ---- Cross-lane suffix-max of segment maxima (wave32 shuffles) ----
    float incl = run;   // max of this lane's segment
    #pragma unroll
    for (int d = 1; d < 32; d <<= 1) {
        float other = __shfl_down(incl, (unsigned)d, 32);
        if (lane + d < 32) incl = fmaxf(incl, other);
    }
    float excl = __shfl_down(incl, 1u, 32);     // suffix max of lanes strictly above
    if (lane == 31) excl = -__builtin_inff();

    // ---- Combine and write results back to padded LDS ----
    #pragma unroll
    for (int i = 0; i < SEG; ++i) smem[segBase + i] = fmaxf(e[i], excl);

    asm volatile("s_wait_dscnt 0" ::: "memory");  // LDS stores visible to async engine

    // ---- Async DMA: LDS -> global (coalesced 16B/lane) ----
    float* gout = out + rowOff;
    #pragma unroll
    for (int k = 0; k < 8; ++k) {
        const int g   = 4 * (lane + 32 * k);
        const int seg = g >> 5;
        const int w   = g & 31;
        unsigned lds_addr =
            (unsigned)(unsigned long long)(&smem[waveBase + (unsigned)(SEG_STRIDE * seg + w)]);
        unsigned long long gaddr = (unsigned long long)(gout + g);
        asm volatile("global_store_async_from_lds_b128 %0, %1, off"
                     :: "v"(gaddr), "v"(lds_addr) : "memory");
    }
    // ASYNCcnt drained by the implicit wait-idle at s_endpgm.
}

extern "C" void kernel_launch(void* const* d_in, const int* in_sizes, int n_in,
                              void* d_out, int out_size, void* d_ws, size_t ws_size,
                              hipStream_t stream) {
    const float* x   = (const float*)d_in[0];
    float*       out = (float*)d_out;

    const int rows   = in_sizes[0] / ROW_W;                       // 32 * 1024 = 32768
    const int blocks = (rows + WAVES_PER_BLOCK - 1) / WAVES_PER_BLOCK;  // 4096

    RightLeftMaxPooling_kernel<<<blocks, 256, 0, stream>>>(x, out, rows);
}